// NeighborsAttentionLayer_22505628631323
// MI455X (gfx1250) — compile-verified
//
#include <hip/hip_runtime.h>

// Problem constants (match reference)
#define KNBR 32      // neighbors
#define ND   128     // NODE_DIM
#define ED   64      // EDGE_DIM
#define NH   8       // heads
#define HD   32      // head dim
#define OUTD 256     // OUT_DIM
#define SLOPE 0.02f

// Padded LDS strides (floats): conflict-free & 16B-aligned rows
#define NDP  132     // node tile row stride
#define EDP  68      // edge tile row stride
#define FP   33      // f_n / f_e row stride

typedef __attribute__((ext_vector_type(2))) float v2f;
typedef __attribute__((ext_vector_type(8))) float v8f;

__device__ __forceinline__ float leaky(float v) { return v > 0.f ? v : SLOPE * v; }

// softmax across the 32 lanes of a wave32 (value per lane)
__device__ __forceinline__ float softmax32(float v) {
    float m = v;
    #pragma unroll
    for (int o = 16; o > 0; o >>= 1) m = fmaxf(m, __shfl_xor(m, o, 32));
    float e = __expf(v - m);
    float s = e;
    #pragma unroll
    for (int o = 16; o > 0; o >>= 1) s += __shfl_xor(s, o, 32);
    return e / s;
}

__global__ __launch_bounds__(256)
void neigh_attn_kernel(const float* __restrict__ x,
                       const float* __restrict__ node_x,
                       const float* __restrict__ edge_x,
                       const float* __restrict__ Wn, const float* __restrict__ bn,
                       const float* __restrict__ We, const float* __restrict__ be,
                       const float* __restrict__ Wx, const float* __restrict__ bx,
                       const float* __restrict__ Wv, const float* __restrict__ bv,
                       const float* __restrict__ W2, const float* __restrict__ b2,
                       float* __restrict__ out)
{
    __shared__ float sNode[KNBR * NDP];   // 32 x 128 (padded)
    __shared__ float sEdge[KNBR * EDP];   // 32 x 64  (padded)
    __shared__ float sX[ND];
    __shared__ float sFn[NH * KNBR * FP]; // per-head f_n [32 x 32] (padded)
    __shared__ float sFe[NH * KNBR * FP];
    __shared__ float sQ[NH * HD];
    __shared__ float sCoef[NH * KNBR];

    const int n   = blockIdx.x;
    const int tid = threadIdx.x;

    // ---------------- stage 1: stage node data in LDS (whole block) ---------
    {
        const float4* gn = (const float4*)(node_x + (size_t)n * KNBR * ND);
        #pragma unroll
        for (int i = tid; i < (KNBR * ND) / 4; i += 256) {
            const int r = i >> 5, c = i & 31;                 // 32 float4 per row
            *(float4*)&sNode[r * NDP + c * 4] = gn[i];
        }
        const float4* ge = (const float4*)(edge_x + (size_t)n * KNBR * ED);
        #pragma unroll
        for (int i = tid; i < (KNBR * ED) / 4; i += 256) {
            const int r = i >> 4, c = i & 15;                 // 16 float4 per row
            *(float4*)&sEdge[r * EDP + c * 4] = ge[i];
        }
        if (tid < ND / 4)
            ((float4*)sX)[tid] = ((const float4*)(x + (size_t)n * ND))[tid];
    }
    __syncthreads();

    // ---------------- self branch: out[:, 0:256] ----------------------------
    {
        float acc = b2[tid];
        #pragma unroll 8
        for (int d = 0; d < ND; ++d) acc = fmaf(sX[d], W2[d * OUTD + tid], acc);
        out[(size_t)n * (2 * OUTD) + tid] = fmaxf(acc, 0.f);
    }

    // ---------------- per-head work: wave h owns head h ---------------------
    const int h    = tid >> 5;
    const int l    = tid & 31;
    const int half = l >> 4;   // WMMA lane-half
    const int lidx = l & 15;

    float* fnH = sFn + h * (KNBR * FP);
    float* feH = sFe + h * (KNBR * FP);

    // f_n = sNode @ Wn[h] + bn[h]   (32x128 @ 128x32, f32 WMMA 16x16x4)
    {
        const float* Bw = Wn + (size_t)h * ND * HD;
        v8f a00 = {}, a01 = {}, a10 = {}, a11 = {};
        #pragma unroll 2
        for (int kk = 0; kk < ND; kk += 4) {
            const int k0 = kk + 2 * half;
            v2f a0, a1, b0, b1;
            a0.x = sNode[lidx * NDP + k0];          a0.y = sNode[lidx * NDP + k0 + 1];
            a1.x = sNode[(16 + lidx) * NDP + k0];   a1.y = sNode[(16 + lidx) * NDP + k0 + 1];
            b0.x = Bw[k0 * HD + lidx];              b0.y = Bw[(k0 + 1) * HD + lidx];
            b1.x = Bw[k0 * HD + 16 + lidx];         b1.y = Bw[(k0 + 1) * HD + 16 + lidx];
            a00 = __builtin_amdgcn_wmma_f32_16x16x4_f32(false, a0, false, b0, (short)0, a00, false, false);
            a01 = __builtin_amdgcn_wmma_f32_16x16x4_f32(false, a0, false, b1, (short)0, a01, false, false);
            a10 = __builtin_amdgcn_wmma_f32_16x16x4_f32(false, a1, false, b0, (short)0, a10, false, false);
            a11 = __builtin_amdgcn_wmma_f32_16x16x4_f32(false, a1, false, b1, (short)0, a11, false, false);
        }
        const float bi0 = bn[h * HD + lidx];
        const float bi1 = bn[h * HD + 16 + lidx];
        #pragma unroll
        for (int j = 0; j < 8; ++j) {
            const int m0 = half * 8 + j;               // D layout: row m0 / m0+16
            fnH[m0 * FP + lidx]             = a00[j] + bi0;
            fnH[m0 * FP + 16 + lidx]        = a01[j] + bi1;
            fnH[(16 + m0) * FP + lidx]      = a10[j] + bi0;
            fnH[(16 + m0) * FP + 16 + lidx] = a11[j] + bi1;
        }
    }

    // f_e = sEdge @ We[h] + be[h]   (32x64 @ 64x32)
    {
        const float* Bw = We + (size_t)h * ED * HD;
        v8f a00 = {}, a01 = {}, a10 = {}, a11 = {};
        #pragma unroll 2
        for (int kk = 0; kk < ED; kk += 4) {
            const int k0 = kk + 2 * half;
            v2f a0, a1, b0, b1;
            a0.x = sEdge[lidx * EDP + k0];          a0.y = sEdge[lidx * EDP + k0 + 1];
            a1.x = sEdge[(16 + lidx) * EDP + k0];   a1.y = sEdge[(16 + lidx) * EDP + k0 + 1];
            b0.x = Bw[k0 * HD + lidx];              b0.y = Bw[(k0 + 1) * HD + lidx];
            b1.x = Bw[k0 * HD + 16 + lidx];         b1.y = Bw[(k0 + 1) * HD + 16 + lidx];
            a00 = __builtin_amdgcn_wmma_f32_16x16x4_f32(false, a0, false, b0, (short)0, a00, false, false);
            a01 = __builtin_amdgcn_wmma_f32_16x16x4_f32(false, a0, false, b1, (short)0, a01, false, false);
            a10 = __builtin_amdgcn_wmma_f32_16x16x4_f32(false, a1, false, b0, (short)0, a10, false, false);
            a11 = __builtin_amdgcn_wmma_f32_16x16x4_f32(false, a1, false, b1, (short)0, a11, false, false);
        }
        const float bi0 = be[h * HD + lidx];
        const float bi1 = be[h * HD + 16 + lidx];
        #pragma unroll
        for (int j = 0; j < 8; ++j) {
            const int m0 = half * 8 + j;
            feH[m0 * FP + lidx]             = a00[j] + bi0;
            feH[m0 * FP + 16 + lidx]        = a01[j] + bi1;
            feH[(16 + m0) * FP + lidx]      = a10[j] + bi0;
            feH[(16 + m0) * FP + 16 + lidx] = a11[j] + bi1;
        }
    }

    // q[h] = leaky(x @ Wx[h] + bx[h])   (lane l computes output channel l)
    {
        const float* Wxh = Wx + (size_t)h * ND * HD;
        float fx = bx[h * HD + l];
        #pragma unroll 8
        for (int d = 0; d < ND; ++d) fx = fmaf(sX[d], Wxh[d * HD + l], fx);
        sQ[h * HD + l] = leaky(fx);
    }

    // logits + dual softmax (lane l == neighbor k)
    {
        float ln = 0.f, le = 0.f;
        #pragma unroll 8
        for (int o = 0; o < HD; ++o) {
            const float qo = sQ[h * HD + o];
            ln = fmaf(qo, leaky(fnH[l * FP + o]), ln);
            le = fmaf(qo, leaky(feH[l * FP + o]), le);
        }
        const float scale = 0.17677669529663687f;   // 1/sqrt(32)
        const float cn = softmax32(ln * scale);
        const float ce = softmax32(le * scale);
        sCoef[h * KNBR + l] = softmax32(cn + ce);
    }

    // v = leaky(cat(f_n, f_e) @ Wv + bv); then values = coef . v; relu; store
    {
        v8f a00 = {}, a01 = {}, a10 = {}, a11 = {};
        #pragma unroll 2
        for (int cc = 0; cc < 2 * HD; cc += 4) {
            const int c0 = cc + 2 * half;
            const float* src; int cl;
            if (c0 < HD) { src = fnH; cl = c0; } else { src = feH; cl = c0 - HD; }
            v2f a0, a1, b0, b1;
            a0.x = src[lidx * FP + cl];          a0.y = src[lidx * FP + cl + 1];
            a1.x = src[(16 + lidx) * FP + cl];   a1.y = src[(16 + lidx) * FP + cl + 1];
            b0.x = Wv[c0 * HD + lidx];           b0.y = Wv[(c0 + 1) * HD + lidx];
            b1.x = Wv[c0 * HD + 16 + lidx];      b1.y = Wv[(c0 + 1) * HD + 16 + lidx];
            a00 = __builtin_amdgcn_wmma_f32_16x16x4_f32(false, a0, false, b0, (short)0, a00, false, false);
            a01 = __builtin_amdgcn_wmma_f32_16x16x4_f32(false, a0, false, b1, (short)0, a01, false, false);
            a10 = __builtin_amdgcn_wmma_f32_16x16x4_f32(false, a1, false, b0, (short)0, a10, false, false);
            a11 = __builtin_amdgcn_wmma_f32_16x16x4_f32(false, a1, false, b1, (short)0, a11, false, false);
        }
        const float bv0 = bv[lidx];
        const float bv1 = bv[16 + lidx];
        float p0 = 0.f, p1 = 0.f;
        #pragma unroll
        for (int j = 0; j < 8; ++j) {
            const int m0 = half * 8 + j;
            const float c_lo = sCoef[h * KNBR + m0];
            const float c_hi = sCoef[h * KNBR + 16 + m0];
            p0 = fmaf(c_lo, leaky(a00[j] + bv0), p0);
            p1 = fmaf(c_lo, leaky(a01[j] + bv1), p1);
            p0 = fmaf(c_hi, leaky(a10[j] + bv0), p0);
            p1 = fmaf(c_hi, leaky(a11[j] + bv1), p1);
        }
        p0 += __shfl_xor(p0, 16, 32);   // fold the two M-halves (same output col)
        p1 += __shfl_xor(p1, 16, 32);
        if (half == 0) {
            float* o = out + (size_t)n * (2 * OUTD) + OUTD + h * HD;
            o[lidx]      = fmaxf(p0, 0.f);
            o[16 + lidx] = fmaxf(p1, 0.f);
        }
    }
}

extern "C" void kernel_launch(void* const* d_in, const int* in_sizes, int n_in,
                              void* d_out, int out_size, void* d_ws, size_t ws_size,
                              hipStream_t stream) {
    const float* x      = (const float*)d_in[0];
    const float* node_x = (const float*)d_in[1];
    const float* edge_x = (const float*)d_in[2];
    const float* Wn     = (const float*)d_in[3];
    const float* bn     = (const float*)d_in[4];
    const float* We     = (const float*)d_in[5];
    const float* be     = (const float*)d_in[6];
    const float* Wx     = (const float*)d_in[7];
    const float* bx     = (const float*)d_in[8];
    const float* Wv     = (const float*)d_in[9];
    const float* bv     = (const float*)d_in[10];
    const float* W2     = (const float*)d_in[11];
    const float* b2     = (const float*)d_in[12];
    float* out = (float*)d_out;

    const int n = in_sizes[0] / ND;   // 50000 nodes
    neigh_attn_kernel<<<n, 256, 0, stream>>>(x, node_x, edge_x,
                                             Wn, bn, We, be, Wx, bx,
                                             Wv, bv, W2, b2, out);
}